// TrilinearInterpolation_82171314307385
// MI455X (gfx1250) — compile-verified
//
#include <hip/hip_runtime.h>
#include <hip/hip_bf16.h>

// 3D LUT trilinear interpolation for MI455X (gfx1250).
// lut: [1,3,33,33,33] fp32, x: [8,3,1024,1024] fp32 in [0,1] -> out [8,3,1024,1024] fp32.
//
// Strategy:
//  - repack_rg: interleave LUT channels 0,1 as float2 per (r,g,b) cell into d_ws
//    (35937 cells -> padded 36864 cells = 294912 B).
//  - lut_apply: stage the rg-pack into LDS (fits in CDNA5's 320KB WGP LDS) using
//    gfx1250 async direct-to-LDS b128 copies (ASYNCcnt + s_wait_asynccnt), then
//    per pixel do 8x ds_load_b64 (r,g corners) + 8x b32 gathers of the blue plane
//    (143KB, L2/WGP$-resident). x/out streamed with non-temporal float4 ops.

typedef float  floatx4 __attribute__((ext_vector_type(4)));
typedef float  floatx2 __attribute__((ext_vector_type(2)));

#define LUT_D      33
#define LUT_D2     1089          // 33*33
#define NCELL      35937         // 33^3
#define IMG_HW     (1024*1024)
#define IMG_B      8
#define THREADS    512
#define LDS_BYTES  294912        // 512 threads * 16B * 36 iters (>= NCELL*8, <= 320KB)
#define LDS_CELLS  (LDS_BYTES/8) // 36864

// ---------------------------------------------------------------- repack ----
__global__ void repack_rg_kernel(const float* __restrict__ lut,
                                 floatx2* __restrict__ ws) {
    int i = blockIdx.x * blockDim.x + threadIdx.x;
    if (i >= LDS_CELLS) return;
    floatx2 v = {0.0f, 0.0f};
    if (i < NCELL) {
        v.x = lut[i];            // channel 0 plane
        v.y = lut[NCELL + i];    // channel 1 plane
    }
    ws[i] = v;
}

// ------------------------------------------------------------ interp core ---
__device__ __forceinline__ void corner_acc(const floatx2* __restrict__ lds2,
                                           const float*   __restrict__ Bp,
                                           int cell, float w,
                                           float& o0, float& o1, float& o2) {
    floatx2 rg = lds2[cell];     // ds_load_b64
    float   bb = Bp[cell];       // global_load_b32 (L2/WGP$ resident)
    o0 = fmaf(w, rg.x, o0);
    o1 = fmaf(w, rg.y, o1);
    o2 = fmaf(w, bb,   o2);
}

__device__ __forceinline__ void lut3d_pixel(float r, float g, float b,
                                            const floatx2* __restrict__ lds2,
                                            const float*   __restrict__ Bp,
                                            float& o0, float& o1, float& o2) {
    float vr = fminf(fmaxf(r, 0.0f), 1.0f) * (float)(LUT_D - 1);
    float vg = fminf(fmaxf(g, 0.0f), 1.0f) * (float)(LUT_D - 1);
    float vb = fminf(fmaxf(b, 0.0f), 1.0f) * (float)(LUT_D - 1);
    int r0 = (int)vr, g0 = (int)vg, b0 = (int)vb;   // vr,vg,vb >= 0 -> trunc == floor
    float fr = vr - (float)r0;
    float fg = vg - (float)g0;
    float fb = vb - (float)b0;
    int dr = (r0 < LUT_D - 1) ? LUT_D2 : 0;
    int dg = (g0 < LUT_D - 1) ? LUT_D  : 0;
    int db = (b0 < LUT_D - 1) ? 1      : 0;
    int base = r0 * LUT_D2 + g0 * LUT_D + b0;
    float wr0 = 1.0f - fr, wg0 = 1.0f - fg, wb0 = 1.0f - fb;

    o0 = 0.0f; o1 = 0.0f; o2 = 0.0f;
    corner_acc(lds2, Bp, base,              wr0 * wg0 * wb0, o0, o1, o2);
    corner_acc(lds2, Bp, base + dr,         fr  * wg0 * wb0, o0, o1, o2);
    corner_acc(lds2, Bp, base + dg,         wr0 * fg  * wb0, o0, o1, o2);
    corner_acc(lds2, Bp, base + dr + dg,    fr  * fg  * wb0, o0, o1, o2);
    corner_acc(lds2, Bp, base + db,         wr0 * wg0 * fb , o0, o1, o2);
    corner_acc(lds2, Bp, base + dr + db,    fr  * wg0 * fb , o0, o1, o2);
    corner_acc(lds2, Bp, base + dg + db,    wr0 * fg  * fb , o0, o1, o2);
    corner_acc(lds2, Bp, base + dr + dg+db, fr  * fg  * fb , o0, o1, o2);
}

// ------------------------------------------------------------- main kernel --
__global__ __launch_bounds__(THREADS, 1)
void lut_apply_kernel(const float* __restrict__ x,
                      const float* __restrict__ lut,
                      const unsigned char* __restrict__ ws_rg,
                      float* __restrict__ out) {
    extern __shared__ __align__(16) unsigned char smem[];

    // LDS byte offset of smem: low 32 bits of the flat shared-aperture address.
    unsigned lds_base = (unsigned)(unsigned long long)(smem);

    // Stage the rg-packed LUT into LDS with CDNA5 async direct-to-LDS copies.
    // 294912 B / 16 B = 18432 chunks; 512 threads -> exactly 36 per thread.
    for (int i = threadIdx.x; i < LDS_BYTES / 16; i += THREADS) {
        unsigned loff = lds_base + (unsigned)(i * 16);
        const unsigned char* gp = ws_rg + (size_t)i * 16;
        asm volatile("global_load_async_to_lds_b128 %0, %1, off"
                     :: "v"(loff), "v"(gp)
                     : "memory");
    }
    asm volatile("s_wait_asynccnt 0" ::: "memory");
    __syncthreads();

    const floatx2* lds2 = (const floatx2*)smem;
    const float*   Bp   = lut + 2 * NCELL;   // blue output-channel plane (143KB, L2-hot)

    const int H4 = IMG_HW / 4;               // float4 groups per plane
    const int P4 = IMG_B * H4;               // total float4 groups
    const int stride = gridDim.x * blockDim.x;

    for (int g4 = blockIdx.x * blockDim.x + threadIdx.x; g4 < P4; g4 += stride) {
        int b  = g4 / H4;
        int h4 = g4 - b * H4;
        const floatx4* xb = (const floatx4*)(x + (size_t)b * 3 * IMG_HW);
        floatx4 R  = __builtin_nontemporal_load(&xb[h4]);
        floatx4 G  = __builtin_nontemporal_load(&xb[h4 + H4]);
        floatx4 Bv = __builtin_nontemporal_load(&xb[h4 + 2 * H4]);

        float o0a, o1a, o2a, o0b, o1b, o2b, o0c, o1c, o2c, o0d, o1d, o2d;
        lut3d_pixel(R.x, G.x, Bv.x, lds2, Bp, o0a, o1a, o2a);
        lut3d_pixel(R.y, G.y, Bv.y, lds2, Bp, o0b, o1b, o2b);
        lut3d_pixel(R.z, G.z, Bv.z, lds2, Bp, o0c, o1c, o2c);
        lut3d_pixel(R.w, G.w, Bv.w, lds2, Bp, o0d, o1d, o2d);

        floatx4 O0 = {o0a, o0b, o0c, o0d};
        floatx4 O1 = {o1a, o1b, o1c, o1d};
        floatx4 O2 = {o2a, o2b, o2c, o2d};

        floatx4* ob = (floatx4*)(out + (size_t)b * 3 * IMG_HW);
        __builtin_nontemporal_store(O0, &ob[h4]);
        __builtin_nontemporal_store(O1, &ob[h4 + H4]);
        __builtin_nontemporal_store(O2, &ob[h4 + 2 * H4]);
    }
}

// ------------------------------------------- fallback (no LDS / small ws) ---
__global__ __launch_bounds__(256, 2)
void lut_apply_global_kernel(const float* __restrict__ x,
                             const float* __restrict__ lut,
                             float* __restrict__ out) {
    const int H4 = IMG_HW / 4;
    const int P4 = IMG_B * H4;
    const int stride = gridDim.x * blockDim.x;
    const float* R0 = lut;
    const float* G0 = lut + NCELL;
    const float* B0 = lut + 2 * NCELL;
    for (int g4 = blockIdx.x * blockDim.x + threadIdx.x; g4 < P4; g4 += stride) {
        int b  = g4 / H4;
        int h4 = g4 - b * H4;
        const floatx4* xb = (const floatx4*)(x + (size_t)b * 3 * IMG_HW);
        floatx4 R  = xb[h4];
        floatx4 G  = xb[h4 + H4];
        floatx4 Bv = xb[h4 + 2 * H4];
        float rr[4] = {R.x, R.y, R.z, R.w};
        float gg[4] = {G.x, G.y, G.z, G.w};
        float bb[4] = {Bv.x, Bv.y, Bv.z, Bv.w};
        float o0[4], o1[4], o2[4];
        #pragma unroll
        for (int k = 0; k < 4; ++k) {
            float vr = fminf(fmaxf(rr[k], 0.0f), 1.0f) * 32.0f;
            float vg = fminf(fmaxf(gg[k], 0.0f), 1.0f) * 32.0f;
            float vb = fminf(fmaxf(bb[k], 0.0f), 1.0f) * 32.0f;
            int r0 = (int)vr, g0i = (int)vg, b0i = (int)vb;
            float fr = vr - r0, fg = vg - g0i, fb = vb - b0i;
            int dr = (r0  < 32) ? LUT_D2 : 0;
            int dg = (g0i < 32) ? LUT_D  : 0;
            int db = (b0i < 32) ? 1      : 0;
            int base = r0 * LUT_D2 + g0i * LUT_D + b0i;
            float wr0 = 1.f - fr, wg0 = 1.f - fg, wb0 = 1.f - fb;
            float a0 = 0.f, a1 = 0.f, a2 = 0.f;
            int cells[8] = {base, base+dr, base+dg, base+dr+dg,
                            base+db, base+dr+db, base+dg+db, base+dr+dg+db};
            float w[8] = {wr0*wg0*wb0, fr*wg0*wb0, wr0*fg*wb0, fr*fg*wb0,
                          wr0*wg0*fb,  fr*wg0*fb,  wr0*fg*fb,  fr*fg*fb};
            #pragma unroll
            for (int c = 0; c < 8; ++c) {
                a0 = fmaf(w[c], R0[cells[c]], a0);
                a1 = fmaf(w[c], G0[cells[c]], a1);
                a2 = fmaf(w[c], B0[cells[c]], a2);
            }
            o0[k] = a0; o1[k] = a1; o2[k] = a2;
        }
        floatx4 O0 = {o0[0], o0[1], o0[2], o0[3]};
        floatx4 O1 = {o1[0], o1[1], o1[2], o1[3]};
        floatx4 O2 = {o2[0], o2[1], o2[2], o2[3]};
        floatx4* ob = (floatx4*)(out + (size_t)b * 3 * IMG_HW);
        __builtin_nontemporal_store(O0, &ob[h4]);
        __builtin_nontemporal_store(O1, &ob[h4 + H4]);
        __builtin_nontemporal_store(O2, &ob[h4 + 2 * H4]);
    }
}

// ------------------------------------------------------------------ launch --
extern "C" void kernel_launch(void* const* d_in, const int* in_sizes, int n_in,
                              void* d_out, int out_size, void* d_ws, size_t ws_size,
                              hipStream_t stream) {
    const float* lut = (const float*)d_in[0];   // [1,3,33,33,33]
    const float* x   = (const float*)d_in[1];   // [8,3,1024,1024]
    float* out = (float*)d_out;

    if (d_ws != nullptr && ws_size >= (size_t)LDS_BYTES) {
        floatx2* ws = (floatx2*)d_ws;
        repack_rg_kernel<<<(LDS_CELLS + 255) / 256, 256, 0, stream>>>(lut, ws);
        // 256 workgroups x 512 threads; 288KB LDS -> one WG per WGP.
        lut_apply_kernel<<<256, THREADS, LDS_BYTES, stream>>>(
            x, lut, (const unsigned char*)d_ws, out);
    } else {
        lut_apply_global_kernel<<<2048, 256, 0, stream>>>(x, lut, out);
    }
}